// ParameterGrouping_55473797595919
// MI455X (gfx1250) — compile-verified
//
#include <hip/hip_runtime.h>
#include <hip/hip_bf16.h>
#include <math.h>

typedef __attribute__((ext_vector_type(2))) float v2f;
typedef __attribute__((ext_vector_type(8))) float v8f;

#define NDIM 512
#define DDIM 512
#define JBLK 8   // pairs (fixed i, consecutive j) handled per wave in kernel 2

// ---------------------------------------------------------------------------
// Kernel 1: left = E @ W1[:D],  right = E @ W1[D:] + b1   (both [512,512] f32)
// One wave32 per 16x16 output tile, native f32 WMMA (16x16x4), K=512.
// Global wave id g in [0, 2048): g>>10 selects left/right, low bits pick tile.
// ---------------------------------------------------------------------------
__global__ __launch_bounds__(256)
void pair_gemm_f32_wmma(const float* __restrict__ E,
                        const float* __restrict__ W1,
                        const float* __restrict__ b1,
                        float* __restrict__ ws /* left | right */) {
    const int lane = threadIdx.x & 31;
    const int wave = threadIdx.x >> 5;
    const int g    = blockIdx.x * 8 + wave;      // 0..2047
    const int w    = g >> 10;                    // 0 = left, 1 = right
    const int tile = g & 1023;                   // 0..1023
    const int tr   = tile >> 5;                  // tile row (M/16)
    const int tc   = tile & 31;                  // tile col (N/16)

    const int mn   = lane & 15;                  // M index (A) / N index (B)
    const int kk   = (lane >> 4) << 1;           // 0 for lanes 0-15, 2 for 16-31

    // A: row (tr*16 + mn) of E, walking K. 8-byte aligned (kk even, k0 mult of 4).
    const float* Arow = E + (tr * 16 + mn) * DDIM;
    // B: column (tc*16 + mn) of W1 part w, walking K (rows of W1).
    const float* Bcol = W1 + (size_t)w * DDIM * DDIM + (tc * 16 + mn);

    v8f c = {};
    #pragma unroll 4
    for (int k0 = 0; k0 < DDIM; k0 += 4) {
        // A fragment 16x4: lanes 0-15 -> A[m][k0+0..1], lanes 16-31 -> A[m][k0+2..3]
        v2f a = *reinterpret_cast<const v2f*>(Arow + k0 + kk);
        // B fragment 4x16: lanes 0-15 -> B[k0+0..1][n], lanes 16-31 -> B[k0+2..3][n]
        v2f b;
        b.x = Bcol[(size_t)(k0 + kk)     * DDIM];
        b.y = Bcol[(size_t)(k0 + kk + 1) * DDIM];
        c = __builtin_amdgcn_wmma_f32_16x16x4_f32(
                /*neg_a=*/false, a, /*neg_b=*/false, b,
                /*c_mod=*/(short)0, c, /*reuse_a=*/false, /*reuse_b=*/false);
    }

    // C/D layout: VGPR r -> row r (lanes 0-15) or row r+8 (lanes 16-31), col = lane&15
    const int col     = tc * 16 + mn;
    const int rowBase = tr * 16 + ((lane >> 4) << 3);
    const float bias  = w ? b1[col] : 0.0f;
    float* dst = ws + (size_t)w * NDIM * DDIM;
    #pragma unroll
    for (int r = 0; r < 8; ++r)
        dst[(size_t)(rowBase + r) * DDIM + col] = c[r] + bias;
}

// ---------------------------------------------------------------------------
// Kernel 2: wave handles pairs (i, j0..j0+JBLK-1), i<=j only.
//   h = left[i] + right[j]; LayerNorm; exact GELU; s = sigmoid(h.W2 + b2);
//   out[i,j] = out[j,i] = s.
// left row, gamma, beta, W2 are register-resident across the JBLK pairs, so
// per-pair L2 traffic is just the 2KB right row.
// ---------------------------------------------------------------------------
__global__ __launch_bounds__(256)
void pair_ln_gelu_dot(const float* __restrict__ ws,
                      const float* __restrict__ gamma,
                      const float* __restrict__ beta,
                      const float* __restrict__ W2,
                      const float* __restrict__ b2,
                      float* __restrict__ out) {
    const int lane = threadIdx.x & 31;
    const int wave = threadIdx.x >> 5;
    const int p    = blockIdx.x * 8 + wave;          // 0 .. N*(N/JBLK)-1
    const int i    = p >> 6;                         // p / (N/JBLK)
    const int j0   = (p & 63) * JBLK;                // start of j block
    if (j0 + JBLK - 1 < i) return;                   // whole block below diagonal

    const float* Lrow = ws + (size_t)i * DDIM;
    const float* Rbase = ws + (size_t)NDIM * DDIM;

    // Register-resident reused vectors (16 elements per lane each).
    float L[16], gam[16], bet[16], w2[16];
    #pragma unroll
    for (int t = 0; t < 16; ++t) {
        const int k = lane + (t << 5);
        L[t]   = Lrow[k];
        gam[t] = gamma[k];
        bet[t] = beta[k];
        w2[t]  = W2[k];
    }
    const float bias2 = b2[0];

    for (int jj = 0; jj < JBLK; ++jj) {
        const int j = j0 + jj;
        if (j < i) continue;                         // uniform across the wave
        const float* R = Rbase + (size_t)j * DDIM;

        float h[16];
        float s = 0.0f;
        #pragma unroll
        for (int t = 0; t < 16; ++t) {
            h[t] = L[t] + R[lane + (t << 5)];
            s += h[t];
        }
        #pragma unroll
        for (int off = 16; off > 0; off >>= 1) s += __shfl_xor(s, off, 32);
        const float mu = s * (1.0f / (float)DDIM);

        float v = 0.0f;
        #pragma unroll
        for (int t = 0; t < 16; ++t) {
            const float d = h[t] - mu;
            v += d * d;
        }
        #pragma unroll
        for (int off = 16; off > 0; off >>= 1) v += __shfl_xor(v, off, 32);
        const float rs = rsqrtf(v * (1.0f / (float)DDIM) + 1e-5f);

        float acc = 0.0f;
        #pragma unroll
        for (int t = 0; t < 16; ++t) {
            const float g  = (h[t] - mu) * rs * gam[t] + bet[t];
            // exact GELU: 0.5*g*(1 + erf(g/sqrt(2)))
            const float ge = 0.5f * g * (1.0f + erff(g * 0.70710678118654752440f));
            acc += ge * w2[t];
        }
        #pragma unroll
        for (int off = 16; off > 0; off >>= 1) acc += __shfl_xor(acc, off, 32);

        if (lane == 0) {
            const float z  = acc + bias2;
            const float sg = 1.0f / (1.0f + expf(-z));
            out[(size_t)i * NDIM + j] = sg;
            out[(size_t)j * NDIM + i] = sg;
        }
    }
}

extern "C" void kernel_launch(void* const* d_in, const int* in_sizes, int n_in,
                              void* d_out, int out_size, void* d_ws, size_t ws_size,
                              hipStream_t stream) {
    const float* E     = (const float*)d_in[0]; // [512, 512]
    const float* W1    = (const float*)d_in[1]; // [1024, 512]
    const float* b1    = (const float*)d_in[2]; // [512]
    const float* gamma = (const float*)d_in[3]; // [512]
    const float* beta  = (const float*)d_in[4]; // [512]
    const float* W2    = (const float*)d_in[5]; // [512, 1]
    const float* b2    = (const float*)d_in[6]; // [1]
    float* out = (float*)d_out;                 // [512, 512]
    float* ws  = (float*)d_ws;                  // left [512*512] | right [512*512]

    // Kernel 1: 2048 tile-waves (left+right), 8 waves per 256-thread block.
    pair_gemm_f32_wmma<<<256, 256, 0, stream>>>(E, W1, b1, ws);
    // Kernel 2: one wave per (i, 8-wide j block): 512*64 waves / 8 per block.
    pair_ln_gelu_dot<<<(NDIM * (NDIM / JBLK)) / 8, 256, 0, stream>>>(
        ws, gamma, beta, W2, b2, out);
}